// Selector_27092653703861
// MI455X (gfx1250) — compile-verified
//
#include <hip/hip_runtime.h>
#include <hip/hip_bf16.h>

typedef __attribute__((ext_vector_type(2))) float v2f;
typedef __attribute__((ext_vector_type(4))) float v4f;
typedef __attribute__((ext_vector_type(8))) float v8f;

#define N_ITEMS 8192
#define D_HALF  64
#define DIN     128
#define HID     128

// ---------------------------------------------------------------------------
// Kernel 1: fused MLP + softmax -> per-item (w_img, w_txt)
// One block = 16 items (one WMMA M-tile). 8 waves, wave w owns hidden cols
// [16w, 16w+16). K-loop: 32 x V_WMMA_F32_16X16X4_F32 (exact fp32 GEMM).
// ---------------------------------------------------------------------------
__global__ __launch_bounds__(256) void mlp_softmax_weights(
    const float* __restrict__ img, const float* __restrict__ txt,
    const float* __restrict__ W1,  const float* __restrict__ b1,
    const float* __restrict__ W2,  const float* __restrict__ b2,
    float* __restrict__ w_img, float* __restrict__ w_txt)
{
    __shared__ float h_s[16][HID];

    const int tid  = threadIdx.x;
    const int wave = tid >> 5;          // 0..7
    const int lane = tid & 31;
    const int half = lane >> 4;         // 0 or 1
    const int l16  = lane & 15;
    const int m0   = blockIdx.x * 16;   // item block
    const int mrow = m0 + l16;          // A-matrix row for this lane
    const int ncol = wave * 16 + l16;   // hidden column for B/C/D on this lane

    v8f acc = {};
    for (int k = 0; k < DIN; k += 4) {
        const int kbase = k + half * 2;
        v2f a, b;
#pragma unroll
        for (int v = 0; v < 2; ++v) {
            const int kk = kbase + v;
            // A: lane l -> A[l%16][(l/16)*2 + v]   (ISA 16x4 f32 layout)
            a[v] = (kk < D_HALF) ? img[mrow * D_HALF + kk]
                                 : txt[mrow * D_HALF + (kk - D_HALF)];
            // B: lane l -> B[(l/16)*2 + v][l%16]
            b[v] = W1[kk * HID + ncol];
        }
        // D = A(16x4) x B(4x16) + C  -- exact fp32
        acc = __builtin_amdgcn_wmma_f32_16x16x4_f32(
            /*neg_a=*/false, a, /*neg_b=*/false, b,
            /*c_mod=*/(short)0, acc, /*reuse_a=*/false, /*reuse_b=*/false);
    }

    // bias + relu, spill h tile to LDS (C/D layout: row r + half*8, col ncol)
    const float bias = b1[ncol];
#pragma unroll
    for (int r = 0; r < 8; ++r) {
        const int m = r + half * 8;
        float h = acc[r] + bias;
        h_s[m][ncol] = h > 0.0f ? h : 0.0f;
    }
    __syncthreads();

    // 2-logit head + softmax: 16 threads, one item each (256 MACs apiece)
    if (tid < 16) {
        float l0 = b2[0], l1 = b2[1];
        for (int n = 0; n < HID; ++n) {
            const float h = h_s[tid][n];
            l0 += h * W2[n * 2 + 0];
            l1 += h * W2[n * 2 + 1];
        }
        const float mx  = fmaxf(l0, l1);
        const float e0  = expf(l0 - mx);
        const float e1  = expf(l1 - mx);
        const float inv = 1.0f / (e0 + e1);
        w_img[m0 + tid] = e0 * inv;
        w_txt[m0 + tid] = e1 * inv;
    }
}

// ---------------------------------------------------------------------------
// Kernel 2: out[i,j] = w_img[j]*image_adj[i,j] + w_txt[j]*text_adj[i,j]
// 768 MB of streaming traffic -> HBM-bound (~33us @ 23.3 TB/s).
// v4f (float4) per thread, exact mapping; NT hints on the streams (zero
// reuse), cached loads for the 64KB weight arrays (reused by every row).
// ---------------------------------------------------------------------------
__global__ __launch_bounds__(256) void blend_adj(
    const v4f* __restrict__ ia, const v4f* __restrict__ ta,
    const v4f* __restrict__ wi4, const v4f* __restrict__ wt4,
    v4f* __restrict__ out)
{
    const size_t idx = (size_t)blockIdx.x * 256u + threadIdx.x; // float4 index
    const int j4 = (int)(idx & (size_t)(N_ITEMS / 4 - 1));      // column group

    const v4f a  = __builtin_nontemporal_load(&ia[idx]);
    const v4f b  = __builtin_nontemporal_load(&ta[idx]);
    const v4f wi = wi4[j4];
    const v4f wt = wt4[j4];

    const v4f o = wi * a + wt * b;
    __builtin_nontemporal_store(o, &out[idx]);
}

extern "C" void kernel_launch(void* const* d_in, const int* in_sizes, int n_in,
                              void* d_out, int out_size, void* d_ws, size_t ws_size,
                              hipStream_t stream) {
    const float* img = (const float*)d_in[0];
    const float* txt = (const float*)d_in[1];
    const float* ia  = (const float*)d_in[2];
    const float* ta  = (const float*)d_in[3];
    const float* W1  = (const float*)d_in[4];
    const float* b1  = (const float*)d_in[5];
    const float* W2  = (const float*)d_in[6];
    const float* b2  = (const float*)d_in[7];

    float* w_img = (float*)d_ws;            // 8192 floats
    float* w_txt = w_img + N_ITEMS;         // 8192 floats (64KB total)

    mlp_softmax_weights<<<N_ITEMS / 16, 256, 0, stream>>>(
        img, txt, W1, b1, W2, b2, w_img, w_txt);

    const size_t n4 = (size_t)N_ITEMS * N_ITEMS / 4;   // 16,777,216 float4
    blend_adj<<<(unsigned)(n4 / 256), 256, 0, stream>>>(
        (const v4f*)ia, (const v4f*)ta,
        (const v4f*)w_img, (const v4f*)w_txt, (v4f*)d_out);
}